// KNNEdge_44667659879031
// MI455X (gfx1250) — compile-verified
//
#include <hip/hip_runtime.h>

// KNNEdge for MI455X (gfx1250, wave32).
// Distance tiles via V_WMMA_F32_16X16X4_F32 (homogeneous packing), per-wave
// top-10 selection with early-out insertion, residue/eps filters, packed output.

#define GRAPHS        8
#define KNN           10
#define MIN_RES_DIST  5
#define EPS2          1e-20f     // (1e-10)^2

#define WAVES_PER_BLOCK 8
#define ROWS_PER_WAVE   16
#define ROWS_PER_BLOCK  (WAVES_PER_BLOCK * ROWS_PER_WAVE)   // 128

typedef __attribute__((ext_vector_type(2))) float v2f;
typedef __attribute__((ext_vector_type(8))) float v8f;

__global__ __launch_bounds__(256) void KNNEdge_44667659879031_kernel(
    const float* __restrict__ pos,          // [N][3]
    const int*   __restrict__ atom2residue, // [N]
    float*       __restrict__ out,          // edge_list[N*K*3] ++ valid[N*K] ++ num_relation[1]
    int nPerGraph, int nTotal)
{
  // Per-wave LDS: 16x16 distance tile + merge scratch. 28 KB / block.
  __shared__ float tile[WAVES_PER_BLOCK][16][16];
  __shared__ float mD[WAVES_PER_BLOCK][32][KNN];
  __shared__ int   mI[WAVES_PER_BLOCK][32][KNN];

  const int tid  = threadIdx.x;
  const int wave = tid >> 5;          // wave32
  const int lane = tid & 31;
  const int half = lane >> 4;         // 0: lanes 0-15, 1: lanes 16-31
  const int l15  = lane & 15;

  // Rows are contiguous per graph; ROWS_PER_BLOCK divides nPerGraph.
  const int blockRow0 = blockIdx.x * ROWS_PER_BLOCK;         // global row base
  const int g         = blockRow0 / nPerGraph;
  const int graphBase = g * nPerGraph;
  const int rowLoc0   = (blockRow0 - graphBase) + wave * ROWS_PER_WAVE;
  const int rowLoc    = rowLoc0 + l15;                       // local row this lane owns
  const int growNode  = graphBase + rowLoc;                  // global query node id

  // ---- A operand (fixed for whole column sweep) ----
  // ISA 32-bit A 16x4 layout: lanes 0-15 -> (K0,K1), lanes 16-31 -> (K2,K3).
  // Row i packed as (x, y, z, ||x||^2).
  const float ax  = pos[growNode * 3 + 0];
  const float ay  = pos[growNode * 3 + 1];
  const float az  = pos[growNode * 3 + 2];
  const float axn = ax * ax + ay * ay + az * az;
  v2f A;
  A.x = half ? az  : ax;
  A.y = half ? axn : ay;

  // ---- per-lane top-K (sorted ascending by d2); 2 lanes per row ----
  float bd[KNN];
  int   bi[KNN];
#pragma unroll
  for (int t = 0; t < KNN; ++t) { bd[t] = 3.4e38f; bi[t] = -1; }

  const int nChunks = nPerGraph >> 4;     // 16 columns per chunk
  for (int ch = 0; ch < nChunks; ++ch) {
    const int cbase = ch << 4;

    // ---- B operand: col j packed as (-2x, -2y, -2z, 1) ----
    // B 4x16 row-striped: VGPR0 = rows K0(l0-15)/K1(l16-31), VGPR1 = K2/K3.
    const int gcol = graphBase + cbase + l15;
    const float cx = pos[gcol * 3 + 0];
    const float cy = pos[gcol * 3 + 1];
    const float cz = pos[gcol * 3 + 2];
    const float cxn = cx * cx + cy * cy + cz * cz;   // xn_j, per-lane column add
    v2f B;
    B.x = half ? (-2.0f * cy) : (-2.0f * cx);
    B.y = half ? 1.0f         : (-2.0f * cz);

    if (ch + 4 < nChunks) {
      // stream-ahead prefetch of the B columns (global_prefetch_b8)
      __builtin_prefetch(&pos[(size_t)(gcol + 64) * 3], 0, 1);
    }

    // D[i][j] = ||x_i||^2 - 2 x_i . x_j   (16x16 f32 accum, EXEC all-ones here)
    v8f C = {};
    v8f D = __builtin_amdgcn_wmma_f32_16x16x4_f32(
        /*neg_a=*/false, A, /*neg_b=*/false, B,
        /*c_mod=*/(short)0, C, /*reuse_a=*/false, /*reuse_b=*/false);

    // C/D layout: VGPR v holds row (v + 8*half), column = l15. Add xn_j per lane.
#pragma unroll
    for (int v = 0; v < 8; ++v)
      tile[wave][v + 8 * half][l15] = D[v] + cxn;

    __builtin_amdgcn_wave_barrier();
    asm volatile("s_wait_dscnt 0x0" ::: "memory");   // wave-local DS ordering

    // ---- selection: lane pair (l15, l15+16) owns row l15; split 8+8 cols ----
    const int c0 = half * 8;
#pragma unroll
    for (int c = 0; c < 8; ++c) {
      const int jloc = cbase + c0 + c;
      const float d  = tile[wave][l15][c0 + c];
      if (jloc == rowLoc) continue;                  // no self loops
      if (d < bd[KNN - 1]) {                         // early-out vs current worst
        float cd = d; int ci = jloc;
#pragma unroll
        for (int s = 0; s < KNN; ++s) {
          const bool le = cd < bd[s];
          const float td = bd[s]; const int ti = bi[s];
          bd[s] = le ? cd : bd[s];
          bi[s] = le ? ci : bi[s];
          cd = le ? td : cd;
          ci = le ? ti : ci;
        }
      }
    }
    __builtin_amdgcn_wave_barrier();
  }

  // ---- merge the two half-lists of each row through LDS ----
#pragma unroll
  for (int t = 0; t < KNN; ++t) {
    mD[wave][lane][t] = bd[t];
    mI[wave][lane][t] = bi[t];
  }
  __builtin_amdgcn_wave_barrier();
  asm volatile("s_wait_dscnt 0x0" ::: "memory");

  if (lane < 16) {
    float fd[KNN];
    int   fi[KNN];
    int pa = 0, pb = 0;
#pragma unroll
    for (int t = 0; t < KNN; ++t) {
      const float da = mD[wave][lane][pa];
      const float db = mD[wave][lane + 16][pb];
      const int   ia = mI[wave][lane][pa];
      const int   ib = mI[wave][lane + 16][pb];
      const bool takeA = (da < db) || (da == db && ia <= ib);
      fd[t] = takeA ? da : db;
      fi[t] = takeA ? ia : ib;
      pa += takeA ? 1 : 0;
      pb += takeA ? 0 : 1;
    }

    const int   ro = atom2residue[growNode];
    float* outV = out + (size_t)nTotal * KNN * 3;
#pragma unroll
    for (int t = 0; t < KNN; ++t) {
      const int nin = graphBase + fi[t];
      const int ri  = atom2residue[nin];
      int rd = ri - ro; rd = rd < 0 ? -rd : rd;
      // eps filter on exact coordinate difference (matches reference norm test)
      const float dx = pos[nin * 3 + 0] - ax;
      const float dy = pos[nin * 3 + 1] - ay;
      const float dz = pos[nin * 3 + 2] - az;
      const bool ok = (rd >= MIN_RES_DIST) && (dx * dx + dy * dy + dz * dz >= EPS2);

      const size_t e = (size_t)growNode * KNN + t;
      out[e * 3 + 0] = ok ? (float)nin      : -1.0f;
      out[e * 3 + 1] = ok ? (float)growNode : -1.0f;
      out[e * 3 + 2] = ok ? 0.0f            : -1.0f;
      outV[e]        = ok ? 1.0f : 0.0f;
    }
  }

  if (blockIdx.x == 0 && tid == 0)
    out[(size_t)nTotal * KNN * 4] = 1.0f;   // num_relation = 1
}

extern "C" void kernel_launch(void* const* d_in, const int* in_sizes, int n_in,
                              void* d_out, int out_size, void* d_ws, size_t ws_size,
                              hipStream_t stream) {
  const float* pos = (const float*)d_in[0];       // node_position [N*3]
  // d_in[1] = node2graph (uniform graph sizes assumed by the reference; unused)
  const int* a2r   = (const int*)d_in[2];         // atom2residue [N]

  const int N    = in_sizes[1];                   // total nodes
  const int nPer = N / GRAPHS;                    // 4096
  const int blocks = N / ROWS_PER_BLOCK;          // 256

  KNNEdge_44667659879031_kernel<<<blocks, 256, 0, stream>>>(
      pos, a2r, (float*)d_out, nPer, N);
}